// SpatialTransformer_5248450036346
// MI455X (gfx1250) — compile-verified
//
#include <hip/hip_runtime.h>

typedef __attribute__((ext_vector_type(2))) float v2f;
typedef __attribute__((ext_vector_type(8))) float v8f;

#define IW 512
#define IH 512
#define NCH 3
#define HW (IH * IW)

// One block (256 threads = 8 waves) per (batch, output row).
// Each wave covers two 32-pixel segments of the row.
// Coordinates for all 32 lanes of a segment come from ONE v_wmma_f32_16x16x4_f32:
//   A(16x4): row0=[t00,t01,t02,0] row1=[t10,t11,t12,0]
//            row8=[0,t01,t02,t00] row9=[0,t11,t12,t10]
//   B(4x16): col n = [x_t(s+n), y_t, 1, x_t(s+16+n)]
//   => lane L (0..31): d[0] = xs(pixel s+L), d[1] = ys(pixel s+L)
__global__ __launch_bounds__(256) void spatial_transform_kernel(
    const float* __restrict__ X, const float* __restrict__ theta,
    float* __restrict__ out) {
  const int i = blockIdx.x;  // output row
  const int b = blockIdx.y;  // batch
  const int lane = threadIdx.x & 31;
  const int wave = threadIdx.x >> 5;

  const float t00 = theta[b * 6 + 0];
  const float t01 = theta[b * 6 + 1];
  const float t02 = theta[b * 6 + 2];
  const float t10 = theta[b * 6 + 3];
  const float t11 = theta[b * 6 + 4];
  const float t12 = theta[b * 6 + 5];

  // A matrix, 32-bit 16x4 layout: lane m(<16): vgpr0=A[m][0], vgpr1=A[m][1];
  //                               lane m+16 : vgpr0=A[m][2], vgpr1=A[m][3].
  float a0 = 0.0f, a1 = 0.0f;
  if (lane == 0)  { a0 = t00; a1 = t01; }
  if (lane == 1)  { a0 = t10; a1 = t11; }
  if (lane == 8)  {           a1 = t01; }
  if (lane == 9)  {           a1 = t11; }
  if (lane == 16) { a0 = t02;           }
  if (lane == 17) { a0 = t12;           }
  if (lane == 24) { a0 = t02; a1 = t00; }
  if (lane == 25) { a0 = t12; a1 = t10; }
  v2f A;
  A[0] = a0;
  A[1] = a1;

  const float dg = 2.0f / (float)(IW - 1);  // linspace(-1,1,512) step
  const float yt = -1.0f + dg * (float)i;

  const float* Xb = X + (size_t)b * NCH * HW;
  float* Ob = out + (size_t)b * NCH * HW + (size_t)i * IW;

#pragma unroll
  for (int it = 0; it < 2; ++it) {
    const int s = (wave * 2 + it) * 32;  // segment base within the row
    const int j = s + lane;              // this lane's output column
    const float xt = -1.0f + dg * (float)j;

    // B matrix, 32-bit 4x16 layout: lane n(<16): vgpr0=B[0][n], vgpr1=B[1][n];
    //                               lane n+16 : vgpr0=B[2][n], vgpr1=B[3][n].
    v2f Bm;
    if (lane < 16) {
      Bm[0] = xt;   // x_t(s+n)
      Bm[1] = yt;   // y_t(i)
    } else {
      Bm[0] = 1.0f; // constant 1
      Bm[1] = xt;   // x_t(s+16+n)
    }

    v8f c = {};
    v8f d = __builtin_amdgcn_wmma_f32_16x16x4_f32(
        false, A, false, Bm, (short)0, c, false, false);

    const float xs = d[0];
    const float ys = d[1];

    const float x = (xs + 1.0f) * (IW * 0.5f);
    const float y = (ys + 1.0f) * (IH * 0.5f);

    const float fx0 = floorf(x);
    const float fy0 = floorf(y);
    const int x0 = (int)fx0;
    const int y0 = (int)fy0;
    const int x1 = x0 + 1;
    const int y1 = y0 + 1;

    const int x0c = min(max(x0, 0), IW - 1);
    const int x1c = min(max(x1, 0), IW - 1);
    const int y0c = min(max(y0, 0), IH - 1);
    const int y1c = min(max(y1, 0), IH - 1);

    const float x0f = (float)x0c;
    const float x1f = (float)x1c;
    const float y0f = (float)y0c;
    const float y1f = (float)y1c;

    const float wa = (x1f - x) * (y1f - y);
    const float wb = (x1f - x) * (y - y0f);
    const float wc = (x - x0f) * (y1f - y);
    const float wd = (x - x0f) * (y - y0f);

    const int ia = y0c * IW + x0c;
    const int ib = y1c * IW + x0c;
    const int ic = y0c * IW + x1c;
    const int id = y1c * IW + x1c;

#pragma unroll
    for (int ch = 0; ch < NCH; ++ch) {
      const float* p = Xb + ch * HW;
      const float v =
          wa * p[ia] + wb * p[ib] + wc * p[ic] + wd * p[id];
      // Output is write-once / streaming: nontemporal store keeps L2 for the
      // gathered input (whole 100MB input fits in the 192MB L2).
      __builtin_nontemporal_store(v, Ob + ch * HW + j);
    }
  }
}

extern "C" void kernel_launch(void* const* d_in, const int* in_sizes, int n_in,
                              void* d_out, int out_size, void* d_ws,
                              size_t ws_size, hipStream_t stream) {
  const float* X = (const float*)d_in[0];
  const float* theta = (const float*)d_in[1];
  float* out = (float*)d_out;
  const int B = in_sizes[1] / 6;  // 32
  dim3 grid(IH, B);
  spatial_transform_kernel<<<grid, 256, 0, stream>>>(X, theta, out);
}